// HHGNN_79242146611943
// MI455X (gfx1250) — compile-verified
//
#include <hip/hip_runtime.h>
#include <hip/hip_bf16.h>

// ---------------------------------------------------------------------------
// HHGNN on MI455X (gfx1250, wave32). FP32 end-to-end; dense GEMMs use
// V_WMMA_F32_16X16X4_F32 (full-precision f32 WMMA). Everything else is
// bandwidth-bound gather/scatter implemented with wave32 shuffles + atomics.
// ---------------------------------------------------------------------------

typedef __attribute__((ext_vector_type(2))) float v2f;
typedef __attribute__((ext_vector_type(8))) float v8f;

#define HC 192   // H*C = 3*64
#define CC 64
#define SLOPE 0.2f

// ---------------- wave32 helpers ----------------

__device__ __forceinline__ float wave_reduce_add(float v) {
    v += __shfl_xor(v, 16, 32);
    v += __shfl_xor(v, 8, 32);
    v += __shfl_xor(v, 4, 32);
    v += __shfl_xor(v, 2, 32);
    v += __shfl_xor(v, 1, 32);
    return v;
}

// branch-per-sign float atomic max (init is -FLT_MAX = 0xFF7FFFFF)
__device__ __forceinline__ void atomicMaxFloat(float* addr, float v) {
    if (v >= 0.0f) atomicMax((int*)addr, __float_as_int(v));
    else           atomicMin((unsigned int*)addr, __float_as_uint(v));
}

// ---------------- WMMA f32 GEMM: out[M,Nc] = A[M,K] @ B[K,Nc] + bias --------
// One wave computes one 16x16 output tile. M%16==0, Nc%16==0, K%4==0 (holds
// for all three GEMMs here: M=82000, K in {64,192}, Nc in {192,64}).
// A frag (16x4 f32): lane l -> row m=l&15, k-pair base kb=(l>>4)*2, v0=K=kb, v1=K=kb+1
// B frag (4x16 f32): lane l -> col n=l&15, v0=row kb, v1=row kb+1
// C/D (16x16 f32, 8 VGPRs): VGPR r -> row r + 8*(l>>4), col l&15

__global__ void wmma_gemm_bias(const float* __restrict__ A,
                               const float* __restrict__ B,
                               const float* __restrict__ bias,
                               float* __restrict__ out,
                               int M, int K, int Nc) {
    int wave = blockIdx.x * (blockDim.x >> 5) + (threadIdx.x >> 5);
    int lane = threadIdx.x & 31;
    int tilesN = Nc >> 4;
    int totalTiles = (M >> 4) * tilesN;
    if (wave >= totalTiles) return;          // whole-wave uniform: EXEC stays all-1s

    int mt = wave / tilesN;
    int nt = wave - mt * tilesN;

    int m   = lane & 15;
    int kb  = (lane >> 4) << 1;
    int col = nt * 16 + (lane & 15);

    const float* Arow = A + (size_t)(mt * 16 + m) * K;
    const float* Bcol = B + col;

    v8f c = {};
    for (int k0 = 0; k0 < K; k0 += 4) {
        v2f a, b;
        a.x = Arow[k0 + kb];
        a.y = Arow[k0 + kb + 1];
        b.x = Bcol[(size_t)(k0 + kb) * Nc];
        b.y = Bcol[(size_t)(k0 + kb + 1) * Nc];
        c = __builtin_amdgcn_wmma_f32_16x16x4_f32(false, a, false, b,
                                                  (short)0, c, false, false);
    }

    float bv = bias[col];
    int rbase = mt * 16 + ((lane >> 4) << 3);
#pragma unroll
    for (int r = 0; r < 8; ++r)
        out[(size_t)(rbase + r) * Nc + col] = c[r] + bv;
}

// ---------------- hop 1: beta = leaky_relu(<Xp[vertex], att_e>_c) ----------
// wave per incidence; per head h a 64-wide dot reduced across 32 lanes.

__global__ void k_edge_beta(const float* __restrict__ Xp,
                            const float* __restrict__ attE,
                            const int* __restrict__ vertex,
                            float* __restrict__ betaRaw, int nnz) {
    int w = blockIdx.x * (blockDim.x >> 5) + (threadIdx.x >> 5);
    int lane = threadIdx.x & 31;
    if (w >= nnz) return;
    const float* row = Xp + (size_t)vertex[w] * HC;
#pragma unroll
    for (int h = 0; h < 3; ++h) {
        float p = row[h * 64 + lane]      * attE[h * 64 + lane]
                + row[h * 64 + lane + 32] * attE[h * 64 + lane + 32];
        p = wave_reduce_add(p);
        if (lane == 0) {
            float s = p > 0.0f ? p : SLOPE * p;
            betaRaw[w * 3 + h] = s;
        }
    }
}

// ---------------- hop 1: per-edge softmax (5 incidences) + aggregation -----
// edges = repeat(arange(E),5) by construction -> incidence [5e,5e+5) per edge.
// One 192-thread block per hyperedge; thread k accumulates element k.

__global__ void k_edge_aggregate(const float* __restrict__ Xp,
                                 const float* __restrict__ betaRaw,
                                 const int* __restrict__ vertex,
                                 float* __restrict__ Xe) {
    int e = blockIdx.x;
    int k = threadIdx.x;               // 0..191
    int h = k >> 6;

    float b[5];
    float mx = -3.402823466e38f;
#pragma unroll
    for (int p = 0; p < 5; ++p) {
        b[p] = betaRaw[(e * 5 + p) * 3 + h];
        mx = fmaxf(mx, b[p]);
    }
    float s = 0.0f;
#pragma unroll
    for (int p = 0; p < 5; ++p) { b[p] = __expf(b[p] - mx); s += b[p]; }
    float inv = 1.0f / s;

    float acc = 0.0f;
#pragma unroll
    for (int p = 0; p < 5; ++p) {
        int v = vertex[e * 5 + p];
        acc += (b[p] * inv) * Xp[(size_t)v * HC + k];
    }
    Xe[(size_t)e * HC + k] = acc;
}

// ---------------- hop 2: class-keyed scores at concat position r -----------
// r walks the concatenation [idx_a; idx_u; idx_i]; the incidence row used for
// the dot is j = perm[r], but the softmax segment key is vertex[r] (faithful
// replication of the reference's gather quirk).

__global__ void k_hop2_score(const float* __restrict__ Xe,
                             const float* __restrict__ attA,
                             const float* __restrict__ attU,
                             const float* __restrict__ attI,
                             const int* __restrict__ idxA,
                             const int* __restrict__ idxU,
                             const int* __restrict__ idxI,
                             const int* __restrict__ vertex,
                             const int* __restrict__ edges,
                             float* __restrict__ score,
                             float* __restrict__ vmax,
                             int nnz, int nA, int nU) {
    int r = blockIdx.x * (blockDim.x >> 5) + (threadIdx.x >> 5);
    int lane = threadIdx.x & 31;
    if (r >= nnz) return;

    const float* att;
    int j;
    if (r < nA)           { j = idxA[r];            att = attA; }
    else if (r < nA + nU) { j = idxU[r - nA];       att = attU; }
    else                  { j = idxI[r - nA - nU];  att = attI; }

    const float* row = Xe + (size_t)edges[j] * HC;
    int vr = vertex[r];
#pragma unroll
    for (int h = 0; h < 3; ++h) {
        float p = row[h * 64 + lane]      * att[h * 64 + lane]
                + row[h * 64 + lane + 32] * att[h * 64 + lane + 32];
        p = wave_reduce_add(p);
        if (lane == 0) {
            float s = p > 0.0f ? p : SLOPE * p;
            score[r * 3 + h] = s;
            atomicMaxFloat(&vmax[vr * 3 + h], s);
        }
    }
}

__global__ void k_hop2_expsum(const int* __restrict__ vertex,
                              float* __restrict__ score,
                              const float* __restrict__ vmax,
                              float* __restrict__ vden, int nnz3) {
    int t = blockIdx.x * blockDim.x + threadIdx.x;
    if (t >= nnz3) return;
    int r = t / 3, h = t - r * 3;
    int vr = vertex[r];
    float e = __expf(score[t] - vmax[vr * 3 + h]);
    score[t] = e;
    atomicAdd(&vden[vr * 3 + h], e);
}

// Xv[vertex[r]] += alpha[r] * Xe[edges[r]]  (original incidence order)
__global__ void k_hop2_scatter(const float* __restrict__ Xe,
                               const float* __restrict__ score,
                               const float* __restrict__ vden,
                               const int* __restrict__ vertex,
                               const int* __restrict__ edges,
                               float* __restrict__ Xv) {
    int r = blockIdx.x;
    int k = threadIdx.x;               // 0..191
    int h = k >> 6;
    int vr = vertex[r];
    float alpha = score[r * 3 + h] / vden[vr * 3 + h];
    float val = alpha * Xe[(size_t)edges[r] * HC + k];
    atomicAdd(&Xv[(size_t)vr * HC + k], val);
}

__global__ void k_relu(float* __restrict__ x, long n) {
    long t = (long)blockIdx.x * blockDim.x + threadIdx.x;
    if (t < n) x[t] = fmaxf(x[t], 0.0f);
}

__global__ void k_fill(float* __restrict__ x, float v, int n) {
    int t = blockIdx.x * blockDim.x + threadIdx.x;
    if (t < n) x[t] = v;
}

// ---------------- host-side layer orchestration ----------------------------

static void run_gemm(const float* A, const float* B, const float* bias,
                     float* out, int M, int K, int Nc, hipStream_t s) {
    int tiles = (M / 16) * (Nc / 16);
    int blocks = (tiles + 7) / 8;              // 8 waves (256 threads) per block
    wmma_gemm_bias<<<blocks, 256, 0, s>>>(A, B, bias, out, M, K, Nc);
}

static void hconv(const float* Xin, int Kin,
                  const float* W, const float* b,
                  const float* attA, const float* attU,
                  const float* attI, const float* attE,
                  const int* vertex, const int* edges,
                  const int* idxA, const int* idxU, const int* idxI,
                  int N, int E, int nnz, int nA, int nU,
                  float* Xp, float* Xe, float* betaRaw, float* score,
                  float* vmax, float* vden, float* Xv, hipStream_t s) {
    // projection: Xp = Xin @ W + b
    run_gemm(Xin, W, b, Xp, N, Kin, HC, s);

    // hop 1
    k_edge_beta<<<(nnz + 7) / 8, 256, 0, s>>>(Xp, attE, vertex, betaRaw, nnz);
    k_edge_aggregate<<<E, HC, 0, s>>>(Xp, betaRaw, vertex, Xe);

    // hop 2 segment softmax over `vertex`
    k_fill<<<(N * 3 + 255) / 256, 256, 0, s>>>(vmax, -3.402823466e38f, N * 3);
    hipMemsetAsync(vden, 0, (size_t)N * 3 * sizeof(float), s);
    k_hop2_score<<<(nnz + 7) / 8, 256, 0, s>>>(Xe, attA, attU, attI,
                                               idxA, idxU, idxI,
                                               vertex, edges, score, vmax,
                                               nnz, nA, nU);
    k_hop2_expsum<<<(nnz * 3 + 255) / 256, 256, 0, s>>>(vertex, score, vmax,
                                                        vden, nnz * 3);

    // scatter to nodes + ReLU (Xv may alias Xin: Xin is dead after the GEMM)
    hipMemsetAsync(Xv, 0, (size_t)N * HC * sizeof(float), s);
    k_hop2_scatter<<<nnz, HC, 0, s>>>(Xe, score, vden, vertex, edges, Xv);
    long n = (long)N * HC;
    k_relu<<<(int)((n + 255) / 256), 256, 0, s>>>(Xv, n);
}

extern "C" void kernel_launch(void* const* d_in, const int* in_sizes, int n_in,
                              void* d_out, int out_size, void* d_ws, size_t ws_size,
                              hipStream_t stream) {
    const float* X     = (const float*)d_in[0];
    const float* W1    = (const float*)d_in[1];
    const float* b1    = (const float*)d_in[2];
    const float* attA1 = (const float*)d_in[3];
    const float* attU1 = (const float*)d_in[4];
    const float* attI1 = (const float*)d_in[5];
    const float* attE1 = (const float*)d_in[6];
    const float* W2    = (const float*)d_in[7];
    const float* b2    = (const float*)d_in[8];
    const float* attA2 = (const float*)d_in[9];
    const float* attU2 = (const float*)d_in[10];
    const float* attI2 = (const float*)d_in[11];
    const float* attE2 = (const float*)d_in[12];
    const float* Wout  = (const float*)d_in[13];
    const float* bout  = (const float*)d_in[14];
    const int* vertex  = (const int*)d_in[15];
    const int* edges   = (const int*)d_in[16];
    const int* idxA    = (const int*)d_in[17];
    const int* idxU    = (const int*)d_in[18];
    const int* idxI    = (const int*)d_in[19];

    const int N   = in_sizes[0] / CC;     // 82000
    const int nnz = in_sizes[15];         // 500000
    const int E   = nnz / 5;              // 100000 (PER_EDGE=5 by construction)
    const int nA  = in_sizes[17];
    const int nU  = in_sizes[18];

    // workspace partition (floats)
    float* ws      = (float*)d_ws;
    float* Xp      = ws;                               // N*192
    float* Xe      = Xp + (size_t)N * HC;              // E*192
    float* betaRaw = Xe + (size_t)E * HC;              // nnz*3
    float* score   = betaRaw + (size_t)nnz * 3;        // nnz*3
    float* vmax    = score + (size_t)nnz * 3;          // N*3
    float* vden    = vmax + (size_t)N * 3;             // N*3
    float* Hbuf    = vden + (size_t)N * 3;             // N*192

    // layer 1: X -> Hbuf
    hconv(X, CC, W1, b1, attA1, attU1, attI1, attE1,
          vertex, edges, idxA, idxU, idxI,
          N, E, nnz, nA, nU,
          Xp, Xe, betaRaw, score, vmax, vden, Hbuf, stream);

    // layer 2: Hbuf -> Hbuf (Hbuf dead after the layer-2 GEMM, reused as Xv)
    hconv(Hbuf, HC, W2, b2, attA2, attU2, attI2, attE2,
          vertex, edges, idxA, idxU, idxI,
          N, E, nnz, nA, nU,
          Xp, Xe, betaRaw, score, vmax, vden, Hbuf, stream);

    // output projection: d_out = Hbuf @ Wout + bout   [N, 64]
    run_gemm(Hbuf, Wout, bout, (float*)d_out, N, HC, CC, stream);
}